// StateMatrixEncoder_27650999452552
// MI455X (gfx1250) — compile-verified
//
#include <hip/hip_runtime.h>
#include <hip/hip_bf16.h>

// StateMatrixEncoder gather+pool for MI455X (gfx1250, wave32).
//
// out[b,l,0,:] = mean_k<4 session[b, sidx[k], stm[b,l,k]-1, :]
// out[b,l,k,:] = session[b, sidx[k], stm[b,l,k]-1, :]   (k = 1..4)
// sidx = {NSESS-1, 0, 1, 2, 3}
//
// Memory-bound gather (no contraction -> WMMA inapplicable). Strategy:
//   * 1 block per (b,l); 128 lanes (4 wave32s), one float4 column per lane
//   * b128 coalesced loads of gathered rows (cache-resident in 192MB L2)
//   * non-temporal b128 stores for the write-once 80 MiB output stream
//   * uniform per-block index loads -> scalarized to SMEM by the compiler

typedef float v4f __attribute__((ext_vector_type(4)));

static constexpr int kB  = 16;
static constexpr int kL  = 512;
static constexpr int kNS = 4;
static constexpr int kS  = 512;
static constexpr int kH  = 512;
static constexpr int kH4 = kH / 4;   // 128 float4 per H-row -> one per lane

__global__ __launch_bounds__(kH4)
void smencoder_gather_pool(const float* __restrict__ sess,
                           const int*   __restrict__ stm,
                           float*       __restrict__ out) {
    const int bl = blockIdx.x;          // b*L + l
    const int b  = bl >> 9;             // bl / kL  (kL == 512)
    const int t  = threadIdx.x;         // float4 column index, 0..127

    const int* __restrict__ stm5 = stm + bl * 5;
    const v4f* __restrict__ s4   = (const v4f*)sess;

    v4f g[5];
#pragma unroll
    for (int k = 0; k < 5; ++k) {
        const int  si  = (k == 0) ? (kNS - 1) : (k - 1);   // {3,0,1,2,3}
        const int  p   = stm5[k] - 1;                      // uniform across block
        const long row = (long)(b * kNS + si) * kS + p;    // row of H floats
        g[k] = s4[row * (long)kH4 + t];                    // global_load_b128
    }

    const v4f pooled = (g[0] + g[1] + g[2] + g[3]) * 0.25f;

    v4f* o4 = (v4f*)out + (long)bl * 5 * kH4 + t;
    __builtin_nontemporal_store(pooled, o4);               // global_store_b128 nt
    __builtin_nontemporal_store(g[1],   o4 + 1 * kH4);
    __builtin_nontemporal_store(g[2],   o4 + 2 * kH4);
    __builtin_nontemporal_store(g[3],   o4 + 3 * kH4);
    __builtin_nontemporal_store(g[4],   o4 + 4 * kH4);
}

extern "C" void kernel_launch(void* const* d_in, const int* in_sizes, int n_in,
                              void* d_out, int out_size, void* d_ws, size_t ws_size,
                              hipStream_t stream) {
    // d_in order per setup_inputs():
    //   [0] utterance_repre    (unused by the reference op)
    //   [1] conversation_repre (unused by the reference op)
    //   [2] session_repre      (B, NSESS, S, H) float32
    //   [3] state_transition_matrix (B, L, 5) integer -> const int*
    //   [4] max_conversation_length (scalar, unused: L fixed at 512)
    const float* sess = (const float*)d_in[2];
    const int*   stm  = (const int*)  d_in[3];
    float*       out  = (float*)d_out;

    (void)in_sizes; (void)n_in; (void)out_size; (void)d_ws; (void)ws_size;

    dim3 grid(kB * kL);   // 8192 blocks, one per (b,l)
    dim3 block(kH4);      // 128 threads = 4 wave32s
    smencoder_gather_pool<<<grid, block, 0, stream>>>(sess, stm, out);
}